// CoupledConvCapsule_12549894439173
// MI455X (gfx1250) — compile-verified
//
#include <hip/hip_runtime.h>

typedef __attribute__((ext_vector_type(2))) float v2f;
typedef __attribute__((ext_vector_type(8))) float v8f;

#define B_    32
#define H_    64
#define W_    64
#define T_INc 8
#define A_INc 16
#define T_OUTc 8
#define A_OUTc 32
#define HC    32
#define WC    32
#define OC    256          // T_OUT * A_OUT
#define SITES 16           // sites per workgroup (WMMA M)
#define ROUTINGS 3

__launch_bounds__(256)
__global__ void capsule_routing_fused(const float* __restrict__ x,
                                      const float* __restrict__ kern,
                                      const float* __restrict__ bias,
                                      float* __restrict__ out)
{
    __shared__ float P_lds[SITES * T_INc * 4];                 // [site][t_in][tap]
    __shared__ float logits_lds[SITES * T_INc * T_OUTc];       // [site][t_in][t_out]
    __shared__ float route_lds[SITES * T_INc * T_OUTc];

    const int tid = threadIdx.x;
    const int lid = tid & 31;
    const int wv  = tid >> 5;        // wave id == owned t_out
    const int g   = lid >> 4;        // half-group (WMMA row block)
    const int n   = lid & 15;        // WMMA column / A row
    const int site0 = blockIdx.x * SITES;

    // ---- stage 0: zero logits; channel-sum the 2x2 patches (reads x once) ----
    for (int i = tid; i < SITES * T_INc * T_OUTc; i += 256) logits_lds[i] = 0.f;

    for (int e = tid; e < SITES * T_INc * 4; e += 256) {
        int site_l = e >> 5;
        int tap    = (e >> 3) & 3;
        int t_in   = e & 7;
        int s  = site0 + site_l;
        int wc = s & (WC - 1);
        int hc = (s >> 5) & (HC - 1);
        int b  = s >> 10;
        int y  = 2 * hc + (tap >> 1);
        int xx = 2 * wc + (tap & 1);
        const float4* p = (const float4*)(x +
            ((((size_t)b * H_ + y) * W_ + xx) * T_INc + t_in) * A_INc);
        float4 v0 = p[0], v1 = p[1], v2 = p[2], v3 = p[3];
        float sum = v0.x + v0.y + v0.z + v0.w + v1.x + v1.y + v1.z + v1.w
                  + v2.x + v2.y + v2.z + v2.w + v3.x + v3.y + v3.z + v3.w;
        P_lds[(site_l * T_INc + t_in) * 4 + tap] = sum;
    }
    __syncthreads();

    // ---- B fragments: per-type 4x256 kernel, this wave's 2x16 column chunks ----
    const int k0 = 2 * g;
    v2f bfrag[T_INc][2];
    #pragma unroll
    for (int t = 0; t < T_INc; ++t)
        #pragma unroll
        for (int h = 0; h < 2; ++h) {
            int col = wv * A_OUTc + h * 16 + n;
            bfrag[t][h].x = kern[(t * 4 + k0 + 0) * OC + col];
            bfrag[t][h].y = kern[(t * 4 + k0 + 1) * OC + col];
        }

    // ---- A fragments: patch sums (16 sites x 4 taps) from LDS ----
    v2f afrag[T_INc];
    #pragma unroll
    for (int t = 0; t < T_INc; ++t) {
        afrag[t].x = P_lds[(n * T_INc + t) * 4 + k0 + 0];
        afrag[t].y = P_lds[(n * T_INc + t) * 4 + k0 + 1];
    }

    // ---- votes via V_WMMA_F32_16X16X4_F32, kept in registers ----
    // acc[t][h] comp j == votes[site0 + j + 8g, t_in=t, t_out=wv, a=h*16+n]
    v8f acc[T_INc][2];
    #pragma unroll
    for (int t = 0; t < T_INc; ++t)
        #pragma unroll
        for (int h = 0; h < 2; ++h) {
            v8f c = {};
            acc[t][h] = __builtin_amdgcn_wmma_f32_16x16x4_f32(
                false, afrag[t], false, bfrag[t][h], (short)0, c, false, false);
        }

    float bi0 = bias[wv * A_OUTc + n];
    float bi1 = bias[wv * A_OUTc + 16 + n];

    float act0[8], act1[8];
    for (int it = 0; it < ROUTINGS; ++it) {
        __syncthreads();
        // route = softmax(logits) over t_out; one thread per (site, t_in)
        if (tid < SITES * T_INc) {
            const float* lg = &logits_lds[tid * T_OUTc];
            float m = lg[0];
            #pragma unroll
            for (int t = 1; t < T_OUTc; ++t) m = fmaxf(m, lg[t]);
            float e[T_OUTc], s = 0.f;
            #pragma unroll
            for (int t = 0; t < T_OUTc; ++t) { e[t] = __expf(lg[t] - m); s += e[t]; }
            float inv = 1.f / s;
            #pragma unroll
            for (int t = 0; t < T_OUTc; ++t) route_lds[tid * T_OUTc + t] = e[t] * inv;
        }
        __syncthreads();

        // preact for this wave's t_out: sum over t_in of route * votes (+bias)
        float pre0[8], pre1[8];
        #pragma unroll
        for (int j = 0; j < 8; ++j) { pre0[j] = bi0; pre1[j] = bi1; }
        #pragma unroll
        for (int t = 0; t < T_INc; ++t)
            #pragma unroll
            for (int j = 0; j < 8; ++j) {
                float r = route_lds[((j + 8 * g) * T_INc + t) * T_OUTc + wv];
                pre0[j] = fmaf(r, acc[t][0][j], pre0[j]);
                pre1[j] = fmaf(r, acc[t][1][j], pre1[j]);
            }

        // squash over A_OUT: 16-lane xor-reduction (stays in half-group) x 2 halves
        #pragma unroll
        for (int j = 0; j < 8; ++j) {
            float s2 = pre0[j] * pre0[j] + pre1[j] * pre1[j];
            s2 += __shfl_xor(s2, 1);
            s2 += __shfl_xor(s2, 2);
            s2 += __shfl_xor(s2, 4);
            s2 += __shfl_xor(s2, 8);
            float sc = sqrtf(s2) / (1.f + s2);   // (v/||v||)*||v||^2/(1+||v||^2)
            act0[j] = pre0[j] * sc;
            act1[j] = pre1[j] * sc;
        }

        // logits += <votes, activation>_A_OUT (skip on final iteration: unused)
        if (it < ROUTINGS - 1) {
            #pragma unroll
            for (int t = 0; t < T_INc; ++t)
                #pragma unroll
                for (int j = 0; j < 8; ++j) {
                    float d = acc[t][0][j] * act0[j] + acc[t][1][j] * act1[j];
                    d += __shfl_xor(d, 1);
                    d += __shfl_xor(d, 2);
                    d += __shfl_xor(d, 4);
                    d += __shfl_xor(d, 8);
                    if (n == t)
                        logits_lds[((j + 8 * g) * T_INc + t) * T_OUTc + wv] += d;
                }
        }
    }

    // ---- write final activation [B,Hc,Wc,T_OUT,A_OUT] ----
    #pragma unroll
    for (int j = 0; j < 8; ++j) {
        size_t s = (size_t)(site0 + j + 8 * g);
        out[(s * T_OUTc + wv) * A_OUTc + n]      = act0[j];
        out[(s * T_OUTc + wv) * A_OUTc + 16 + n] = act1[j];
    }
}

extern "C" void kernel_launch(void* const* d_in, const int* in_sizes, int n_in,
                              void* d_out, int out_size, void* d_ws, size_t ws_size,
                              hipStream_t stream) {
    const float* x    = (const float*)d_in[0];
    const float* kern = (const float*)d_in[1];
    const float* bias = (const float*)d_in[2];
    float* out = (float*)d_out;
    const int n_wg = (B_ * HC * WC) / SITES;   // 2048 workgroups, 16 sites each
    capsule_routing_fused<<<n_wg, 256, 0, stream>>>(x, kern, bias, out);
}